// UpSampler_21225728377361
// MI455X (gfx1250) — compile-verified
//
#include <hip/hip_runtime.h>
#include <hip/hip_bf16.h>

typedef __attribute__((ext_vector_type(2))) float v2f;
typedef __attribute__((ext_vector_type(4))) float v4f;
typedef __attribute__((ext_vector_type(8))) float v8f;

#define KNN 6
#define D 128
#define APITCH 132   // LDS pitch: stride-4-dword rows -> conflict-free b64 A-frag loads
#define MAXB 64

// ---------------- Kernel 1a: init batch bounds ----------------
__global__ void init_bounds_kernel(int* __restrict__ starts, int* __restrict__ ends, int Nc) {
    int t = threadIdx.x;
    if (t < MAXB) { starts[t] = Nc; ends[t] = Nc; }  // empty range by default
}

// ---------------- Kernel 1b: find per-batch [start,end) in sorted batch_coarse ----------------
__global__ void find_bounds_kernel(const int* __restrict__ bc,
                                   int* __restrict__ starts, int* __restrict__ ends, int Nc) {
    int j = blockIdx.x * blockDim.x + threadIdx.x;
    if (j >= Nc) return;
    int b = bc[j] & (MAXB - 1);
    if (j == 0 || (bc[j - 1] & (MAXB - 1)) != b) starts[b] = j;
    if (j == Nc - 1 || (bc[j + 1] & (MAXB - 1)) != b) ends[b] = j + 1;
}

// ---------------- Kernel 2: batch-restricted kNN (K=6) + normalized IDW weights ----------------
__global__ __launch_bounds__(256) void knn_kernel(
    const float* __restrict__ pf, const float* __restrict__ pc,
    const int* __restrict__ bf,
    const int* __restrict__ starts, const int* __restrict__ ends,
    int* __restrict__ oidx, float* __restrict__ ow, int Nf)
{
    int i = blockIdx.x * blockDim.x + threadIdx.x;
    if (i >= Nf) return;

    const float px = pf[i * 3 + 0];
    const float py = pf[i * 3 + 1];
    const float pz = pf[i * 3 + 2];
    const int b = bf[i] & (MAXB - 1);
    const int cs = starts[b];
    const int ce = ends[b];

    float bd[KNN];
    int   bi[KNN];
#pragma unroll
    for (int k = 0; k < KNN; ++k) { bd[k] = 1e10f; bi[k] = k; }

    for (int j = cs; j < ce; ++j) {
        float dx = px - pc[j * 3 + 0];
        float dy = py - pc[j * 3 + 1];
        float dz = pz - pc[j * 3 + 2];
        float d2 = fmaf(dx, dx, fmaf(dy, dy, dz * dz));
        if (d2 < bd[KNN - 1]) {            // strict <: earlier index wins ties (matches top_k)
            bd[KNN - 1] = d2; bi[KNN - 1] = j;
#pragma unroll
            for (int k = KNN - 1; k > 0; --k) {
                if (bd[k] < bd[k - 1]) {
                    float td = bd[k]; bd[k] = bd[k - 1]; bd[k - 1] = td;
                    int   ti = bi[k]; bi[k] = bi[k - 1]; bi[k - 1] = ti;
                }
            }
        }
    }

    float w[KNN]; float s = 0.0f;
#pragma unroll
    for (int k = 0; k < KNN; ++k) { w[k] = 1.0f / fmaxf(bd[k], 1e-16f); s += w[k]; }
    const float inv = 1.0f / s;
#pragma unroll
    for (int k = 0; k < KNN; ++k) {
        oidx[i * KNN + k] = bi[k];
        ow[i * KNN + k]   = w[k] * inv;    // fold normalization into weights
    }
}

// ---------------- Kernel 3: fused interpolation-gather + fp32 WMMA GEMM + bias ----------------
// Block: 256 threads (8 waves). M-tile = 16 fine rows, N = 128, K = 128.
// Wave w computes a 16x16 output tile at columns [16w, 16w+16) via 32 x V_WMMA_F32_16X16X4_F32.
__global__ __launch_bounds__(256) void interp_gemm_kernel(
    const float* __restrict__ xc,      // [Nc, 128]
    const float* __restrict__ lw,      // [128, 128] (K x N)
    const float* __restrict__ lb,      // [128]
    const int* __restrict__ knn_idx,   // [Nf, 6]
    const float* __restrict__ knn_w,   // [Nf, 6] (pre-normalized)
    float* __restrict__ out)           // [Nf, 128]
{
    __shared__ float Atile[16 * APITCH];

    const int tid = threadIdx.x;
    const int m0  = blockIdx.x << 4;

    // --- Stage A-tile: interp[row][c] = sum_k w_k * x_coarse[idx_k][c] ---
    {
        const int row = tid >> 4;           // 0..15 (fine row in tile)
        const int cb  = (tid & 15) << 3;    // 8 consecutive feature cols per thread
        const int fi  = m0 + row;
        float acc[8];
#pragma unroll
        for (int q = 0; q < 8; ++q) acc[q] = 0.0f;
#pragma unroll
        for (int k = 0; k < KNN; ++k) {
            const int   j = knn_idx[fi * KNN + k];
            const float w = knn_w[fi * KNN + k];
            const float* src = xc + (size_t)j * D + cb;
            v4f v0 = *(const v4f*)(src);
            v4f v1 = *(const v4f*)(src + 4);
#pragma unroll
            for (int q = 0; q < 4; ++q) {
                acc[q]     = fmaf(w, v0[q], acc[q]);
                acc[q + 4] = fmaf(w, v1[q], acc[q + 4]);
            }
        }
        float* dstp = &Atile[row * APITCH + cb];
#pragma unroll
        for (int q = 0; q < 8; ++q) dstp[q] = acc[q];
    }
    __syncthreads();

    // --- WMMA: D = A(16x128) x lin_w(128x128-slice) ---
    const int lane = tid & 31;
    const int wave = tid >> 5;
    const int nb   = wave << 4;         // N base of this wave's 16-col tile
    const int hi   = lane >> 4;         // selects K pair {0,1} vs {2,3} within a k-step
    const int ln   = lane & 15;         // M for A-frag, N for B-frag

    v8f c = {};
    const float* arow = &Atile[ln * APITCH];
    const float* bptr = lw + nb + ln;

#pragma unroll 4
    for (int kk = 0; kk < D / 4; ++kk) {
        const int k0 = (kk << 2) + (hi << 1);
        v2f a = *(const v2f*)(arow + k0);              // A[M=ln][k0], A[M=ln][k0+1]
        v2f b;
        b[0] = bptr[(size_t)k0 * D];                    // B[k0]  [N=nb+ln]
        b[1] = bptr[(size_t)(k0 + 1) * D];              // B[k0+1][N=nb+ln]
        c = __builtin_amdgcn_wmma_f32_16x16x4_f32(
                /*neg_a=*/false, a, /*neg_b=*/false, b,
                /*c_mod=*/(short)0, c, /*reuse_a=*/false, /*reuse_b=*/false);
    }

    // --- Bias + store: vgpr r -> M = r + 8*hi, N = nb + ln ---
    const float bias = lb[nb + ln];
    float* op = out + (size_t)(m0 + (hi << 3)) * D + nb + ln;
#pragma unroll
    for (int r = 0; r < 8; ++r)
        op[(size_t)r * D] = c[r] + bias;
}

extern "C" void kernel_launch(void* const* d_in, const int* in_sizes, int n_in,
                              void* d_out, int out_size, void* d_ws, size_t ws_size,
                              hipStream_t stream) {
    const float* x_coarse     = (const float*)d_in[0];
    const float* pos_coarse   = (const float*)d_in[1];
    const float* pos_fine     = (const float*)d_in[2];
    const float* lin_w        = (const float*)d_in[3];
    const float* lin_b        = (const float*)d_in[4];
    const int*   batch_coarse = (const int*)d_in[5];
    const int*   batch_fine   = (const int*)d_in[6];

    const int Nc = in_sizes[1] / 3;
    const int Nf = in_sizes[2] / 3;

    // Workspace layout
    int*   starts  = (int*)d_ws;                 // MAXB ints
    int*   ends    = starts + MAXB;              // MAXB ints
    int*   knn_idx = ends + MAXB;                // Nf*6 ints
    float* knn_w   = (float*)(knn_idx + (size_t)Nf * KNN);
    float* out     = (float*)d_out;

    init_bounds_kernel<<<1, MAXB, 0, stream>>>(starts, ends, Nc);
    find_bounds_kernel<<<(Nc + 255) / 256, 256, 0, stream>>>(batch_coarse, starts, ends, Nc);
    knn_kernel<<<(Nf + 255) / 256, 256, 0, stream>>>(pos_fine, pos_coarse, batch_fine,
                                                     starts, ends, knn_idx, knn_w, Nf);
    interp_gemm_kernel<<<Nf / 16, 256, 0, stream>>>(x_coarse, lin_w, lin_b,
                                                    knn_idx, knn_w, out);
}